// Y_ENCODER_58506044506603
// MI455X (gfx1250) — compile-verified
//
#include <hip/hip_runtime.h>
#include <math.h>

// ---------------------------------------------------------------------------
// GCN 2-layer forward for MI455X (gfx1250, wave32).
//   xn = l2norm(x); h = relu(spmm(xn@W1+b1)); out = softmax(spmm(h@W2+b2))
// GEMMs use V_WMMA_F32_16X16X4_F32 (the fp32 WMMA), SpMM uses f32 atomics
// into L2-resident buffers (all node tensors fit in the 192MB L2).
// ---------------------------------------------------------------------------

typedef __attribute__((ext_vector_type(2))) float v2f;
typedef __attribute__((ext_vector_type(8))) float v8f;

#define NND 100000
#define NED 1600000
#define DK  128      // inner dim for both GEMMs
#define DH  128
#define NC  64

// ------------------------- row L2 normalize --------------------------------
__global__ void gcn_l2norm(const float* __restrict__ x, float* __restrict__ xn) {
  int row  = blockIdx.x * 8 + (threadIdx.x >> 5);
  int lane = threadIdx.x & 31;
  if (row >= NND) return;
  float4 v = ((const float4*)(x + (size_t)row * DH))[lane];
  float s = v.x * v.x + v.y * v.y + v.z * v.z + v.w * v.w;
  #pragma unroll
  for (int o = 16; o > 0; o >>= 1) s += __shfl_xor(s, o, 32);
  float inv = 1.0f / fmaxf(sqrtf(s), 1e-12f);
  float4 r;
  r.x = v.x * inv; r.y = v.y * inv; r.z = v.z * inv; r.w = v.w * inv;
  ((float4*)(xn + (size_t)row * DH))[lane] = r;
}

// ------------------- fp32 WMMA GEMM:  Out = [relu](A) @ W + b --------------
// A: [M x 128] row-major, W: [128 x NCOLS] row-major, Out: [M x NCOLS].
// Block: 256 thr = 8 waves, covers 128 rows. Wave: 16 rows x NCOLS.
// LDS: W transposed, row stride 132 dwords -> conflict-free ds_load_b64.
template <int NCOLS, int RELU_IN>
__global__ void gcn_gemm_wmma(const float* __restrict__ A,
                              const float* __restrict__ W,
                              const float* __restrict__ bias,
                              float* __restrict__ Out, int M) {
  constexpr int NT  = NCOLS / 16;   // accumulator tiles per wave
  constexpr int LDK = DK + 4;       // 132-dword padded row
  extern __shared__ float lds[];    // Wt[NCOLS][LDK]

  const int tid = threadIdx.x;
  // stage W transposed: Wt[n][k] = W[k][n]  (coalesced global reads)
  for (int i = tid; i < DK * NCOLS; i += 256) {
    int k = i / NCOLS, n = i % NCOLS;
    lds[n * LDK + k] = W[i];
  }
  __syncthreads();

  const int wave = tid >> 5;
  const int lane = tid & 31;
  const int lm   = lane & 15;       // sub-lane row/col index
  const int hi   = lane >> 4;       // lane half (selects K pair / M+8)
  const int m0   = blockIdx.x * 128 + wave * 16;

  // A-fragment row for this lane (masked for the grid tail)
  const int mr = m0 + lm;
  const float mask = (mr < M) ? 1.0f : 0.0f;
  const float* arow = A + (size_t)((mr < M) ? mr : 0) * DK;

  v8f c[NT];
  #pragma unroll
  for (int t = 0; t < NT; ++t) c[t] = (v8f)0.0f;

  for (int k0 = 0; k0 < DK; k0 += 4) {
    const int ka = k0 + 2 * hi;     // K index pair held by this lane half
    v2f a = *(const v2f*)(arow + ka);
    if (RELU_IN) { a.x = fmaxf(a.x, 0.0f); a.y = fmaxf(a.y, 0.0f); }
    a = a * mask;
    #pragma unroll
    for (int t = 0; t < NT; ++t) {
      const int n = t * 16 + lm;
      v2f b = *(const v2f*)&lds[n * LDK + ka];
      // D = A(16x4) * B(4x16) + C   -> v_wmma_f32_16x16x4_f32
      c[t] = __builtin_amdgcn_wmma_f32_16x16x4_f32(
          false, a, false, b, (short)0, c[t], false, false);
    }
  }

  // bias add + store (C/D layout: vgpr r, lane half hi -> row m0+r+8*hi, col lm)
  #pragma unroll
  for (int t = 0; t < NT; ++t) {
    const int n = t * 16 + lm;
    const float bz = bias[n];
    #pragma unroll
    for (int r = 0; r < 8; ++r) {
      const int row = m0 + r + 8 * hi;
      if (row < M) Out[(size_t)row * NCOLS + n] = c[t][r] + bz;
    }
  }
}

// ------------------------ SpMM: agg[dst] += w * h[src] ---------------------
__global__ void gcn_spmm128(const float* __restrict__ h, const int* __restrict__ src,
                            const int* __restrict__ dst, const float* __restrict__ w,
                            float* __restrict__ agg) {
  int e = blockIdx.x * 8 + (threadIdx.x >> 5);
  if (e >= NED) return;
  int lane = threadIdx.x & 31;
  int s = src[e], d = dst[e];
  float we = w[e];
  float4 v = ((const float4*)(h + (size_t)s * 128))[lane];
  float* o = agg + (size_t)d * 128 + lane * 4;
  atomicAdd(o + 0, v.x * we);
  atomicAdd(o + 1, v.y * we);
  atomicAdd(o + 2, v.z * we);
  atomicAdd(o + 3, v.w * we);
}

__global__ void gcn_spmm64(const float* __restrict__ h, const int* __restrict__ src,
                           const int* __restrict__ dst, const float* __restrict__ w,
                           float* __restrict__ agg) {
  int e = blockIdx.x * 8 + (threadIdx.x >> 5);
  if (e >= NED) return;
  int lane = threadIdx.x & 31;
  int s = src[e], d = dst[e];
  float we = w[e];
  float2 v = ((const float2*)(h + (size_t)s * 64))[lane];
  float* o = agg + (size_t)d * 64 + lane * 2;
  atomicAdd(o + 0, v.x * we);
  atomicAdd(o + 1, v.y * we);
}

// ------------------------------ row softmax --------------------------------
__global__ void gcn_softmax(const float* __restrict__ z, float* __restrict__ out) {
  int row = blockIdx.x * 8 + (threadIdx.x >> 5);
  if (row >= NND) return;
  int lane = threadIdx.x & 31;
  float2 v = ((const float2*)(z + (size_t)row * NC))[lane];
  float m = fmaxf(v.x, v.y);
  #pragma unroll
  for (int o = 16; o > 0; o >>= 1) m = fmaxf(m, __shfl_xor(m, o, 32));
  float e0 = __expf(v.x - m), e1 = __expf(v.y - m);
  float s = e0 + e1;
  #pragma unroll
  for (int o = 16; o > 0; o >>= 1) s += __shfl_xor(s, o, 32);
  float inv = 1.0f / s;
  float2 r; r.x = e0 * inv; r.y = e1 * inv;
  ((float2*)(out + (size_t)row * NC))[lane] = r;
}

// ---------------------------------------------------------------------------
extern "C" void kernel_launch(void* const* d_in, const int* in_sizes, int n_in,
                              void* d_out, int out_size, void* d_ws, size_t ws_size,
                              hipStream_t stream) {
  (void)in_sizes; (void)n_in; (void)out_size; (void)ws_size;
  const float* x  = (const float*)d_in[0];
  const int*   src = (const int*)d_in[1];
  const int*   dst = (const int*)d_in[2];
  const float* ew = (const float*)d_in[3];
  const float* W1 = (const float*)d_in[4];
  const float* b1 = (const float*)d_in[5];
  const float* W2 = (const float*)d_in[6];
  const float* b2 = (const float*)d_in[7];
  float* out = (float*)d_out;

  float* bufA = (float*)d_ws;                     // 100k x 128 f32 (51.2 MB)
  float* bufB = bufA + (size_t)NND * DH;          // 100k x 128 f32 (51.2 MB)

  const int gemmGrid = (NND + 127) / 128;
  const size_t lds1 = (size_t)DH * (DK + 4) * sizeof(float);  // 128x132 dwords
  const size_t lds2 = (size_t)NC * (DK + 4) * sizeof(float);  //  64x132 dwords

  // 1. xn = l2norm(x)                          -> bufA
  gcn_l2norm<<<(NND + 7) / 8, 256, 0, stream>>>(x, bufA);
  // 2. h = xn @ W1 + b1  (fp32 WMMA)           -> bufB
  gcn_gemm_wmma<128, 0><<<gemmGrid, 256, lds1, stream>>>(bufA, W1, b1, bufB, NND);
  // 3. agg = spmm(h)                            -> bufA (zeroed; xn is dead)
  hipMemsetAsync(bufA, 0, (size_t)NND * 128 * sizeof(float), stream);
  gcn_spmm128<<<(NED + 7) / 8, 256, 0, stream>>>(bufB, src, dst, ew, bufA);
  // 4. h2 = relu(agg) @ W2 + b2  (fp32 WMMA)   -> bufB (h is dead)
  gcn_gemm_wmma<64, 1><<<gemmGrid, 256, lds2, stream>>>(bufA, W2, b2, bufB, NND);
  // 5. pre = spmm(h2)                           -> bufA (zeroed; agg is dead)
  hipMemsetAsync(bufA, 0, (size_t)NND * 64 * sizeof(float), stream);
  gcn_spmm64<<<(NED + 7) / 8, 256, 0, stream>>>(bufB, src, dst, ew, bufA);
  // 6. out = softmax(pre)
  gcn_softmax<<<(NND + 7) / 8, 256, 0, stream>>>(bufA, out);
}